// GraphDownSamplingLayer_76716705841224
// MI455X (gfx1250) — compile-verified
//
#include <hip/hip_runtime.h>
#include <stdint.h>

// Problem constants (fixed by the reference's setup_inputs()).
constexpr int B = 16;
constexpr int F = 128;
constexpr int N = 16384;
constexpr int P = 4096;

constexpr int THREADS = 1024;          // 32 wave32 waves = one full WGP, 8 waves/SIMD
constexpr int SLOTS   = N / THREADS;   // 16 points per thread, kept in VGPRs

// Dynamic LDS layout (floats):
//   [0, 3N)          : point cloud copy (px | py | pz), 192 KB — needs CDNA5's 320KB WGP LDS
//   [3N, 3N+64)      : double-buffered per-wave best value   [2][32]
//   [3N+64, 3N+128)  : double-buffered per-wave best index   [2][32] (int bits)
constexpr int LDS_FLOATS = 3 * N + 128;

__global__ __launch_bounds__(THREADS, 1)
void fps_kernel(const float* __restrict__ pos, int* __restrict__ idx_out)
{
    extern __shared__ float smem[];
    float* px   = smem;                       // [N]
    float* py   = smem + N;                   // [N]
    float* pz   = smem + 2 * N;               // [N]
    float* redv = smem + 3 * N;               // [2][32]
    int*   redi = (int*)(smem + 3 * N + 64);  // [2][32]

    const int b = blockIdx.x;
    const int t = threadIdx.x;
    const float* src = pos + (size_t)b * 3 * N;   // contiguous [3][N] block, 16B aligned

    // ---- gfx1250 async DMA: copy the whole 192KB point block global -> LDS ----
    // LDS address for the async op is workgroup-relative; generic LDS pointers
    // carry that offset in their low 32 bits.
    {
        uint32_t ldsBase = (uint32_t)(uintptr_t)smem;
        uint64_t g = (uint64_t)(uintptr_t)src;
        #pragma unroll
        for (int k = 0; k < (3 * N) / (4 * THREADS); ++k) {   // 12 x B128 per thread
            uint32_t off  = (uint32_t)(t + k * THREADS) * 16u;
            uint32_t lOff = ldsBase + off;
            uint64_t ga   = g + off;
            asm volatile("global_load_async_to_lds_b128 %0, %1, off"
                         :: "v"(lOff), "v"(ga) : "memory");
        }
        asm volatile("s_wait_asynccnt 0" ::: "memory");
    }
    __syncthreads();

    // Cache this thread's 16 points + running min distances in registers.
    float rx[SLOTS], ry[SLOTS], rz[SLOTS], md[SLOTS];
    #pragma unroll
    for (int s = 0; s < SLOTS; ++s) {
        const int i = t + s * THREADS;
        rx[s] = px[i];
        ry[s] = py[i];
        rz[s] = pz[i];
        md[s] = 3.4e38f;
    }

    if (t == 0) idx_out[(size_t)b * P + 0] = 0;   // FPS starts at index 0
    const int wave = t >> 5;
    const int lane = t & 31;

    int cur = 0;   // uniform across the block at all times
    int par = 0;   // partials double-buffer parity
    for (int p = 1; p < P; ++p) {
        // Broadcast read of the current query point (uniform LDS address).
        const float qx = px[cur], qy = py[cur], qz = pz[cur];

        // Per-thread: update min distances, track local argmax.
        // Slot order gives strictly increasing global indices, so strict '>'
        // already implements first-index tie-breaking within the thread.
        float bv = -1.0f;
        int   bi = t;
        #pragma unroll
        for (int s = 0; s < SLOTS; ++s) {
            const float dx = rx[s] - qx;
            const float dy = ry[s] - qy;
            const float dz = rz[s] - qz;
            const float d  = dx * dx + dy * dy + dz * dz;
            const float m  = md[s] < d ? md[s] : d;
            md[s] = m;
            if (m > bv) { bv = m; bi = t + s * THREADS; }
        }

        // wave32 argmax reduction (first-index tie-break, matches jnp.argmax).
        #pragma unroll
        for (int off = 16; off > 0; off >>= 1) {
            const float ov = __shfl_xor(bv, off, 32);
            const int   oi = __shfl_xor(bi, off, 32);
            if (ov > bv || (ov == bv && oi < bi)) { bv = ov; bi = oi; }
        }
        if (lane == 0) { redv[par * 32 + wave] = bv; redi[par * 32 + wave] = bi; }
        __syncthreads();

        // Every wave redundantly reduces the 32 per-wave partials: all lanes of
        // all waves end up with the same winner -> no broadcast, no 2nd barrier.
        // Double-buffering (par) prevents step p+1 writes racing laggard readers.
        float v  = redv[par * 32 + lane];
        int   i2 = redi[par * 32 + lane];
        #pragma unroll
        for (int off = 16; off > 0; off >>= 1) {
            const float ov = __shfl_xor(v, off, 32);
            const int   oi = __shfl_xor(i2, off, 32);
            if (ov > v || (ov == v && oi < i2)) { v = ov; i2 = oi; }
        }
        cur = i2;
        if (t == 0) idx_out[(size_t)b * P + p] = cur;
        par ^= 1;
    }
}

// Gather: d_out = x_out [B][F][P] followed by pos_out [B][3][P].
__global__ void gather_kernel(const float* __restrict__ x,
                              const float* __restrict__ pos,
                              const int* __restrict__ idx,
                              float* __restrict__ out)
{
    const size_t tid = (size_t)blockIdx.x * blockDim.x + threadIdx.x;
    const size_t totalX = (size_t)B * F * P;
    const size_t totalPos = (size_t)B * 3 * P;
    if (tid < totalX) {
        const int p = (int)(tid % P);
        const int f = (int)((tid / P) % F);
        const int b = (int)(tid / ((size_t)F * P));
        const int i = idx[(size_t)b * P + p];
        out[tid] = x[((size_t)b * F + f) * N + i];
    } else if (tid < totalX + totalPos) {
        const size_t r = tid - totalX;
        const int p = (int)(r % P);
        const int c = (int)((r / P) % 3);
        const int b = (int)(r / ((size_t)3 * P));
        const int i = idx[(size_t)b * P + p];
        out[tid] = pos[((size_t)b * 3 + c) * N + i];
    }
}

extern "C" void kernel_launch(void* const* d_in, const int* in_sizes, int n_in,
                              void* d_out, int out_size, void* d_ws, size_t ws_size,
                              hipStream_t stream)
{
    (void)in_sizes; (void)n_in; (void)out_size; (void)ws_size;
    const float* pos = (const float*)d_in[0];   // [B,3,N,1] f32
    const float* x   = (const float*)d_in[1];   // [B,F,N,1] f32
    // d_in[2] is num_points (==4096, compile-time constant here)

    int*   idx = (int*)d_ws;                    // [B*P] selected indices (256 KB)
    float* out = (float*)d_out;

    const size_t shmem = (size_t)LDS_FLOATS * sizeof(float);  // ~192.5 KB < 320 KB
    // Opt in to >64KB dynamic LDS where the runtime requires it (ignore errors).
    (void)hipFuncSetAttribute((const void*)fps_kernel,
                              hipFuncAttributeMaxDynamicSharedMemorySize, (int)shmem);

    fps_kernel<<<B, THREADS, shmem, stream>>>(pos, idx);

    const size_t total = (size_t)B * F * P + (size_t)B * 3 * P;  // 8,585,216
    const int blocks = (int)((total + 255) / 256);
    gather_kernel<<<blocks, 256, 0, stream>>>(x, pos, idx, out);
}